// Pointnet2MSG_manimix_20624432956051
// MI455X (gfx1250) — compile-verified
//
#include <hip/hip_runtime.h>
#include <math.h>

// ---------------- constants ----------------
#define BB 16
#define NN 1024
#define NANG 60
#define PI_F 3.14159274f
#define EPS_F 0.005f
#define SINKIT 30
#define LOGMU (-6.931471806f) /* -log(1024) */

typedef __attribute__((ext_vector_type(2))) float v2f;
typedef __attribute__((ext_vector_type(8))) float v8f;

#if __has_builtin(__builtin_amdgcn_wmma_f32_16x16x4_f32)
#define HAVE_WMMA4 1
#endif

// D = A(16x4,f32) * B(4x16,f32) + C ; one wave, CDNA5 v_wmma_f32_16x16x4_f32
__device__ __forceinline__ v8f wmma4(v2f a, v2f b, v8f c) {
#ifdef HAVE_WMMA4
  return __builtin_amdgcn_wmma_f32_16x16x4_f32(false, a, false, b, (short)0, c,
                                               false, false);
#else
  // shuffle-based fallback with identical lane layout
  int lane = threadIdx.x & 31;
  v8f d = c;
#pragma unroll
  for (int v = 0; v < 8; ++v) {
    int row = v + ((lane >> 4) << 3);
    int col = lane & 15;
    float acc = 0.f;
#pragma unroll
    for (int k = 0; k < 4; ++k) {
      int al = (row & 15) + ((k >= 2) ? 16 : 0);
      float av = (k & 1) ? __shfl(a.y, al, 32) : __shfl(a.x, al, 32);
      int bl = col + ((k >= 2) ? 16 : 0);
      float bv = (k & 1) ? __shfl(b.y, bl, 32) : __shfl(b.x, bl, 32);
      acc += av * bv;
    }
    d[v] += acc;
  }
  return d;
#endif
}

// One MLP layer on a 16-row chunk: s_O(16,Nc) = relu(s_A(16,K) * Wt(K,Nc) + bias)
// A is LDS-resident, Wt streamed from global (L2-resident). One wave per
// 16-column output tile; K stepped by 4 with zero-padded guards.
__device__ __forceinline__ void mlp_layer_wmma(const float* s_A, int K,
                                               const float* __restrict__ Wt,
                                               const float* __restrict__ bias,
                                               int Nc, float* s_O, int lane,
                                               int wid, int nwaves) {
  const int l15 = lane & 15, hi = lane >> 4;
  for (int nt = wid; nt < (Nc >> 4); nt += nwaves) {
    v8f acc = {};
    const int n = nt * 16 + l15;
    for (int k = 0; k < K; k += 4) {
      int kb = k + hi * 2; // lanes 16-31 carry K=+2,+3
      float a0 = (kb < K) ? s_A[l15 * K + kb] : 0.f;
      float a1 = (kb + 1 < K) ? s_A[l15 * K + kb + 1] : 0.f;
      float w0 = (kb < K) ? Wt[(size_t)kb * Nc + n] : 0.f;
      float w1 = (kb + 1 < K) ? Wt[(size_t)(kb + 1) * Nc + n] : 0.f;
      v2f av; av.x = a0; av.y = a1;
      v2f wv; wv.x = w0; wv.y = w1;
      acc = wmma4(av, wv, acc);
    }
    float bv = bias[n];
#pragma unroll
    for (int v = 0; v < 8; ++v)
      s_O[(v + hi * 8) * Nc + n] = fmaxf(acc[v] + bv, 0.f);
  }
}

// ---------------- small utility kernels ----------------
__global__ void k_norms(const float* __restrict__ xyz, float* __restrict__ xx) {
  int t = blockIdx.x * blockDim.x + threadIdx.x;
  if (t >= BB * NN) return;
  const float* p = xyz + (size_t)t * 3;
  xx[t] = p[0] * p[0] + p[1] * p[1] + p[2] * p[2];
}

// ---------------- chamfer angle search (WMMA f32 16x16x4) ----------------
// grid: (64 row-tiles, B, 2*NANG). pass0: rows=x cols=x@R ; pass1: rows=x@R cols=x
// Reflection R is orthogonal -> |x@R| = |x|, so one norm buffer serves both sides.
__global__ __launch_bounds__(32) void k_chamfer(const float* __restrict__ xyz,
                                                const float* __restrict__ xx,
                                                float* __restrict__ cd) {
  const int lane = threadIdx.x;
  const int rt = blockIdx.x;
  const int b = blockIdx.y;
  const int ang = blockIdx.z >> 1;
  const int pass = blockIdx.z & 1;

  const float th = (float)ang / (float)NANG * PI_F;
  const float s = sinf(th), c = cosf(th);
  const float R00 = 1.f - 2.f * s * s, R02 = 2.f * s * c,
              R22 = 1.f - 2.f * c * c;

  const int m = lane & 15;
  const int hi = lane >> 4;

  const float* xb = xyz + (size_t)b * NN * 3;
  const float* xxb = xx + (size_t)b * NN;

  // A fragment: 16 rows
  int r0 = rt * 16 + m;
  float p0 = xb[r0 * 3 + 0], p1 = xb[r0 * 3 + 1], p2 = xb[r0 * 3 + 2];
  float ar0, ar1, ar2;
  if (pass == 0) { ar0 = p0; ar1 = p1; ar2 = p2; }
  else { ar0 = p0 * R00 + p2 * R02; ar1 = p1; ar2 = p0 * R02 + p2 * R22; }
  v2f A;
  A.x = hi ? ar2 : ar0;   // lanes 16-31 hold K=2,3
  A.y = hi ? 0.f : ar1;   // K=3 is the zero pad

  float xa[8];
#pragma unroll
  for (int v = 0; v < 8; ++v) xa[v] = xxb[rt * 16 + v + hi * 8];

  float rmin[8];
#pragma unroll
  for (int v = 0; v < 8; ++v) rmin[v] = 3.0e38f;

  for (int t = 0; t < NN / 16; ++t) {
    int cidx = t * 16 + m;
    float q0 = xb[cidx * 3 + 0], q1 = xb[cidx * 3 + 1], q2 = xb[cidx * 3 + 2];
    float br0, br1, br2;
    if (pass == 0) { br0 = q0 * R00 + q2 * R02; br1 = q1; br2 = q0 * R02 + q2 * R22; }
    else { br0 = q0; br1 = q1; br2 = q2; }
    v2f Bf;
    Bf.x = hi ? br2 : br0;
    Bf.y = hi ? 0.f : br1;
    float xcol = xxb[cidx];
    v8f acc = {};
    acc = wmma4(A, Bf, acc);
#pragma unroll
    for (int v = 0; v < 8; ++v) {
      float d = xa[v] + xcol - 2.f * acc[v];
      rmin[v] = fminf(rmin[v], d);
    }
  }
  // min across the 16 lanes of each half -> row minimum
#pragma unroll
  for (int v = 0; v < 8; ++v) {
    float x = rmin[v];
    x = fminf(x, __shfl_xor(x, 1, 32));
    x = fminf(x, __shfl_xor(x, 2, 32));
    x = fminf(x, __shfl_xor(x, 4, 32));
    x = fminf(x, __shfl_xor(x, 8, 32));
    rmin[v] = x;
  }
  float ssum = 0.f;
#pragma unroll
  for (int v = 0; v < 8; ++v) ssum += rmin[v];
  float other = __shfl(ssum, 16, 32);
  if (lane == 0) atomicAdd(&cd[ang * BB + b], ssum + other);
}

__global__ void k_select_theta(const float* __restrict__ cd,
                               float* __restrict__ thetas) {
  int b = threadIdx.x;
  if (b >= BB) return;
  float best = cd[b];
  int bi = 0;
  for (int a = 1; a < NANG; ++a) {
    float v = cd[a * BB + b];
    if (v < best) { best = v; bi = a; }
  }
  thetas[b] = (float)bi / (float)NANG * PI_F;
}

__global__ void k_minor_rot(const float* __restrict__ xyz,
                            const int* __restrict__ idxm,
                            const float* __restrict__ thetas,
                            float* __restrict__ mrot, float* __restrict__ yy) {
  int t = blockIdx.x * blockDim.x + threadIdx.x;
  if (t >= BB * NN) return;
  int b = t / NN, n = t % NN;
  int mb = idxm[b];
  const float* p = xyz + ((size_t)mb * NN + n) * 3;
  float td = thetas[b] - thetas[mb];
  float cc = cosf(td), ss = sinf(td);
  float o0 = p[0] * cc - p[2] * ss;
  float o1 = p[1];
  float o2 = p[0] * ss + p[2] * cc;
  float* o = mrot + (size_t)t * 3;
  o[0] = o0; o[1] = o1; o[2] = o2;
  yy[t] = o0 * o0 + o1 * o1 + o2 * o2;
}

// ---------------- Sinkhorn (on-the-fly cost, LDS-staged operands) ----------------
__global__ __launch_bounds__(256) void k_sink_f(
    const float* __restrict__ xyz, const float* __restrict__ xx,
    const float* __restrict__ mrot, const float* __restrict__ yy,
    const float* __restrict__ g, float* __restrict__ f) {
  __shared__ float sy[NN * 3];
  __shared__ float syy[NN];
  __shared__ float sg[NN];
  int b = blockIdx.y;
  for (int k = threadIdx.x; k < NN; k += 256) {
    sy[k * 3 + 0] = mrot[((size_t)b * NN + k) * 3 + 0];
    sy[k * 3 + 1] = mrot[((size_t)b * NN + k) * 3 + 1];
    sy[k * 3 + 2] = mrot[((size_t)b * NN + k) * 3 + 2];
    syy[k] = yy[b * NN + k];
    sg[k] = g[b * NN + k];
  }
  __syncthreads();
  int n = blockIdx.x * 256 + threadIdx.x;
  const float* p = xyz + ((size_t)b * NN + n) * 3;
  float p0 = p[0], p1 = p[1], p2 = p[2];
  float xn = xx[b * NN + n];
  float vmax = -3.0e38f, ssum = 0.f;
  for (int mm = 0; mm < NN; ++mm) {
    float cost = xn + syy[mm] -
                 2.f * (p0 * sy[mm * 3] + p1 * sy[mm * 3 + 1] + p2 * sy[mm * 3 + 2]);
    float v = (sg[mm] - cost) * (1.0f / EPS_F);
    if (v > vmax) { ssum = ssum * __expf(vmax - v) + 1.f; vmax = v; }
    else ssum += __expf(v - vmax);
  }
  f[b * NN + n] = EPS_F * (LOGMU - (vmax + __logf(ssum)));
}

__global__ __launch_bounds__(256) void k_sink_g(
    const float* __restrict__ xyz, const float* __restrict__ xx,
    const float* __restrict__ mrot, const float* __restrict__ yy,
    const float* __restrict__ f, float* __restrict__ g) {
  __shared__ float sx[NN * 3];
  __shared__ float sxx[NN];
  __shared__ float sf[NN];
  int b = blockIdx.y;
  for (int k = threadIdx.x; k < NN; k += 256) {
    sx[k * 3 + 0] = xyz[((size_t)b * NN + k) * 3 + 0];
    sx[k * 3 + 1] = xyz[((size_t)b * NN + k) * 3 + 1];
    sx[k * 3 + 2] = xyz[((size_t)b * NN + k) * 3 + 2];
    sxx[k] = xx[b * NN + k];
    sf[k] = f[b * NN + k];
  }
  __syncthreads();
  int mcol = blockIdx.x * 256 + threadIdx.x;
  const float* q = mrot + ((size_t)b * NN + mcol) * 3;
  float q0 = q[0], q1 = q[1], q2 = q[2];
  float yn = yy[b * NN + mcol];
  float vmax = -3.0e38f, ssum = 0.f;
  for (int nn = 0; nn < NN; ++nn) {
    float cost = sxx[nn] + yn -
                 2.f * (q0 * sx[nn * 3] + q1 * sx[nn * 3 + 1] + q2 * sx[nn * 3 + 2]);
    float v = (sf[nn] - cost) * (1.0f / EPS_F);
    if (v > vmax) { ssum = ssum * __expf(vmax - v) + 1.f; vmax = v; }
    else ssum += __expf(v - vmax);
  }
  g[b * NN + mcol] = EPS_F * (LOGMU - (vmax + __logf(ssum)));
}

__global__ __launch_bounds__(256) void k_assign(
    const float* __restrict__ xyz, const float* __restrict__ xx,
    const float* __restrict__ mrot, const float* __restrict__ yy,
    const float* __restrict__ g, int* __restrict__ ass) {
  __shared__ float sy[NN * 3];
  __shared__ float syy[NN];
  __shared__ float sg[NN];
  int b = blockIdx.y;
  for (int k = threadIdx.x; k < NN; k += 256) {
    sy[k * 3 + 0] = mrot[((size_t)b * NN + k) * 3 + 0];
    sy[k * 3 + 1] = mrot[((size_t)b * NN + k) * 3 + 1];
    sy[k * 3 + 2] = mrot[((size_t)b * NN + k) * 3 + 2];
    syy[k] = yy[b * NN + k];
    sg[k] = g[b * NN + k];
  }
  __syncthreads();
  int n = blockIdx.x * 256 + threadIdx.x;
  const float* p = xyz + ((size_t)b * NN + n) * 3;
  float p0 = p[0], p1 = p[1], p2 = p[2];
  float xn = xx[b * NN + n];
  float best = -3.0e38f;
  int bi = 0;
  for (int mm = 0; mm < NN; ++mm) {
    float cost = xn + syy[mm] -
                 2.f * (p0 * sy[mm * 3] + p1 * sy[mm * 3 + 1] + p2 * sy[mm * 3 + 2]);
    float sc = sg[mm] - cost; // f[n] constant over mm, drop it
    if (sc > best) { best = sc; bi = mm; }
  }
  ass[b * NN + n] = bi;
}

__global__ void k_mix(const float* __restrict__ xyz, const int* __restrict__ idxm,
                      const float* __restrict__ mixr, const int* __restrict__ ass,
                      float* __restrict__ out) {
  int t = blockIdx.x * blockDim.x + threadIdx.x;
  if (t >= BB * NN) return;
  int b = t / NN, n = t % NN;
  float r = mixr[b];
  int mb = idxm[b];
  int a = ass[b * NN + n];
  const float* p = xyz + ((size_t)b * NN + n) * 3;
  const float* q = xyz + ((size_t)mb * NN + a) * 3;
  float* o = out + (size_t)t * 3;
  o[0] = p[0] * (1.f - r) + q[0] * r;
  o[1] = p[1] * (1.f - r) + q[1] * r;
  o[2] = p[2] * (1.f - r) + q[2] * r;
}

// ---------------- farthest point sampling ----------------
__global__ __launch_bounds__(256) void k_fps(const float* __restrict__ xyz,
                                             int Np, int npoint,
                                             int* __restrict__ fidx) {
  __shared__ float dist[NN];
  __shared__ float redv[256];
  __shared__ int redi[256];
  __shared__ int s_last;
  __shared__ float lp[3];
  int b = blockIdx.x, tid = threadIdx.x;
  for (int j = tid; j < Np; j += 256) dist[j] = 1e10f;
  if (tid == 0) { fidx[b * npoint] = 0; s_last = 0; }
  __syncthreads();
  for (int it = 1; it < npoint; ++it) {
    if (tid == 0) {
      const float* p = xyz + ((size_t)b * Np + s_last) * 3;
      lp[0] = p[0]; lp[1] = p[1]; lp[2] = p[2];
    }
    __syncthreads();
    float bv = -1.f;
    int bi = 0x7fffffff;
    for (int j = tid; j < Np; j += 256) {
      const float* p = xyz + ((size_t)b * Np + j) * 3;
      float d0 = p[0] - lp[0], d1 = p[1] - lp[1], d2 = p[2] - lp[2];
      float nd = fminf(dist[j], d0 * d0 + d1 * d1 + d2 * d2);
      dist[j] = nd;
      if (nd > bv || (nd == bv && j < bi)) { bv = nd; bi = j; }
    }
    redv[tid] = bv;
    redi[tid] = bi;
    __syncthreads();
    for (int sN = 128; sN > 0; sN >>= 1) {
      if (tid < sN) {
        if (redv[tid + sN] > redv[tid] ||
            (redv[tid + sN] == redv[tid] && redi[tid + sN] < redi[tid])) {
          redv[tid] = redv[tid + sN];
          redi[tid] = redi[tid + sN];
        }
      }
      __syncthreads();
    }
    if (tid == 0) { fidx[b * npoint + it] = redi[0]; s_last = redi[0]; }
    __syncthreads();
  }
}

__global__ void k_gather(const float* __restrict__ src, int Np,
                         const int* __restrict__ fidx, int S,
                         float* __restrict__ dst) {
  int t = blockIdx.x * blockDim.x + threadIdx.x;
  if (t >= BB * S) return;
  int b = t / S;
  int j = fidx[t];
  const float* p = src + ((size_t)b * Np + j) * 3;
  float* o = dst + (size_t)t * 3;
  o[0] = p[0]; o[1] = p[1]; o[2] = p[2];
}

// ---------------- ball query (indices in ascending order, pad with first) ----
__global__ void k_ball(const float* __restrict__ xyz, int Np,
                       const float* __restrict__ nxyz, int S, float r2, int ns,
                       int* __restrict__ out) {
  int t = blockIdx.x * blockDim.x + threadIdx.x;
  if (t >= BB * S) return;
  int b = t / S;
  const float* c = nxyz + (size_t)t * 3;
  float c0 = c[0], c1 = c[1], c2 = c[2];
  int* o = out + (size_t)t * ns;
  int cnt = 0;
  for (int j = 0; j < Np && cnt < ns; ++j) {
    const float* p = xyz + ((size_t)b * Np + j) * 3;
    float d0 = p[0] - c0, d1 = p[1] - c1, d2 = p[2] - c2;
    if (d0 * d0 + d1 * d1 + d2 * d2 < r2) o[cnt++] = j;
  }
  int pad = (cnt > 0) ? o[0] : 0;
  for (; cnt < ns; ++cnt) o[cnt] = pad;
}

// ---- fused group + 3-layer pointwise MLP (WMMA per 16-sample chunk) + max ----
#define GM_THREADS 256
__global__ __launch_bounds__(GM_THREADS) void k_group_mlp(
    const float* __restrict__ xyz, const float* __restrict__ feats, int Cf,
    int Np, const float* __restrict__ nxyz, int S, const int* __restrict__ bidx,
    int ns, const float* __restrict__ W0, const float* __restrict__ b0, int c1,
    const float* __restrict__ W1, const float* __restrict__ b1, int c2,
    const float* __restrict__ W2, const float* __restrict__ b2, int c3,
    float* __restrict__ outf, int Ctot, int coff) {
  __shared__ float s_in[16 * 324]; // input chunk (also reused for layer-3 out)
  __shared__ float s_h1[16 * 256];
  __shared__ float s_h2[16 * 256];
  __shared__ float s_max[256];
  int blk = blockIdx.x;
  int b = blk / S;
  int tid = threadIdx.x;
  int lane = tid & 31, wid = tid >> 5;
  const int nwaves = GM_THREADS / 32;
  int cin = 3 + Cf;
  const float* ctr = nxyz + (size_t)blk * 3;
  float cx = ctr[0], cy = ctr[1], cz = ctr[2];
  for (int t = tid; t < c3; t += GM_THREADS) s_max[t] = -3.0e38f;
  const int* gidx = bidx + (size_t)blk * ns;
  int nchunk = ns / 16;
  for (int ch = 0; ch < nchunk; ++ch) {
    __syncthreads();
    // stage 16 grouped input rows (rel-xyz ++ gathered feats) into LDS
    for (int t = tid; t < 16 * cin; t += GM_THREADS) {
      int s = t / cin, cc = t % cin;
      int j = gidx[ch * 16 + s];
      float v;
      if (cc < 3)
        v = xyz[((size_t)b * Np + j) * 3 + cc] - ((cc == 0) ? cx : ((cc == 1) ? cy : cz));
      else
        v = feats[((size_t)b * Np + j) * Cf + (cc - 3)];
      s_in[s * cin + cc] = v;
    }
    __syncthreads();
    // three 16xKxN WMMA layers (bias + relu fused in fragment store)
    mlp_layer_wmma(s_in, cin, W0, b0, c1, s_h1, lane, wid, nwaves);
    __syncthreads();
    mlp_layer_wmma(s_h1, c1, W1, b1, c2, s_h2, lane, wid, nwaves);
    __syncthreads();
    mlp_layer_wmma(s_h2, c2, W2, b2, c3, s_in, lane, wid, nwaves); // reuse s_in
    __syncthreads();
    // running max over the 16 samples of this chunk
    for (int t = tid; t < c3; t += GM_THREADS) {
      float m = s_max[t];
      for (int s = 0; s < 16; ++s) m = fmaxf(m, s_in[s * c3 + t]);
      s_max[t] = m;
    }
  }
  __syncthreads();
  for (int t = tid; t < c3; t += GM_THREADS)
    outf[(size_t)blk * Ctot + coff + t] = s_max[t];
}

// ---------------- SA3 input concat ----------------
__global__ void k_concat_sa3(const float* __restrict__ nxyz2,
                             const float* __restrict__ feats2,
                             float* __restrict__ out) {
  int t = blockIdx.x * blockDim.x + threadIdx.x;
  if (t >= 2048 * 643) return;
  int row = t / 643, c = t % 643;
  out[t] = (c < 3) ? nxyz2[row * 3 + c] : feats2[(size_t)row * 640 + (c - 3)];
}

// ---------------- generic WMMA GEMM: C = act(A(M,K)*W(K,N) + bias) ----------------
__global__ __launch_bounds__(32) void k_gemm_wmma(
    const float* __restrict__ A, const float* __restrict__ W,
    const float* __restrict__ bias, float* __restrict__ C, int M, int K, int Nc,
    int act) {
  int lane = threadIdx.x;
  int nt = blockIdx.x, mt = blockIdx.y;
  int l15 = lane & 15, hi = lane >> 4;
  int m = mt * 16 + l15;
  int n = nt * 16 + l15;
  int np = (n < Nc) ? n : 0;
  v8f acc = {};
  for (int k = 0; k < K; k += 4) {
    int kb = k + hi * 2; // lanes 16-31 carry K=+2,+3
    float a0 = (m < M && kb < K) ? A[(size_t)m * K + kb] : 0.f;
    float a1 = (m < M && kb + 1 < K) ? A[(size_t)m * K + kb + 1] : 0.f;
    float w0 = (n < Nc && kb < K) ? W[(size_t)kb * Nc + n] : 0.f;
    float w1 = (n < Nc && kb + 1 < K) ? W[(size_t)(kb + 1) * Nc + n] : 0.f;
    // branchless prefetch of upcoming W rows (global_prefetch_b8)
    int kpf = (kb + 32 < K) ? kb + 32 : kb;
    __builtin_prefetch(&W[(size_t)kpf * Nc + np], 0, 0);
    v2f av; av.x = a0; av.y = a1;
    v2f wv; wv.x = w0; wv.y = w1;
    acc = wmma4(av, wv, acc);
  }
#pragma unroll
  for (int v = 0; v < 8; ++v) {
    int r = mt * 16 + v + hi * 8;
    if (r < M && n < Nc) {
      float o = acc[v] + bias[n];
      if (act) o = fmaxf(o, 0.f);
      C[(size_t)r * Nc + n] = o;
    }
  }
}

__global__ void k_maxpool_sa3(const float* __restrict__ h, float* __restrict__ out) {
  int t = blockIdx.x * blockDim.x + threadIdx.x;
  if (t >= BB * 1024) return;
  int b = t / 1024, c = t % 1024;
  float m = -3.0e38f;
  for (int i = 0; i < 128; ++i)
    m = fmaxf(m, h[((size_t)b * 128 + i) * 1024 + c]);
  out[t] = m;
}

// ---------------- host orchestration ----------------
extern "C" void kernel_launch(void* const* d_in, const int* in_sizes, int n_in,
                              void* d_out, int out_size, void* d_ws,
                              size_t ws_size, hipStream_t stream) {
  (void)n_in; (void)out_size; (void)ws_size;
  const float* pc = (const float*)d_in[0];
  const int* idxm = (const int*)d_in[1];
  const float* mixr = (const float*)d_in[2];

  // pytree flatten order detection: sorted keys (fc,sa1,sa2,sa3) vs insertion
  int fcB, sa1B, sa2B, sa3B;
  if (in_sizes[3] == 1024 * 512) { fcB = 3; sa1B = 9; sa2B = 27; sa3B = 45; }
  else { sa1B = 3; sa2B = 21; sa3B = 39; fcB = 45; }
  auto F = [&](int i) { return (const float*)d_in[i]; };

  size_t off = 0;
  auto alloc = [&](size_t bytes) -> void* {
    void* p = (char*)d_ws + off;
    off += (bytes + 255) & ~(size_t)255;
    return p;
  };
  float* w_xx = (float*)alloc((size_t)BB * NN * 4);
  float* w_cd = (float*)alloc((size_t)NANG * BB * 4);
  float* w_th = (float*)alloc((size_t)BB * 4);
  float* w_mrot = (float*)alloc((size_t)BB * NN * 3 * 4);
  float* w_yy = (float*)alloc((size_t)BB * NN * 4);
  float* w_f = (float*)alloc((size_t)BB * NN * 4);
  float* w_g = (float*)alloc((size_t)BB * NN * 4);
  int* w_ass = (int*)alloc((size_t)BB * NN * 4);
  float* w_mix = (float*)alloc((size_t)BB * NN * 3 * 4);
  int* w_fps1 = (int*)alloc((size_t)BB * 512 * 4);
  float* w_nx1 = (float*)alloc((size_t)BB * 512 * 3 * 4);
  int* w_ball = (int*)alloc((size_t)BB * 512 * 128 * 4);
  float* w_ft1 = (float*)alloc((size_t)BB * 512 * 320 * 4);
  int* w_fps2 = (int*)alloc((size_t)BB * 128 * 4);
  float* w_nx2 = (float*)alloc((size_t)BB * 128 * 3 * 4);
  float* w_ft2 = (float*)alloc((size_t)BB * 128 * 640 * 4);
  float* w_s3i = (float*)alloc((size_t)2048 * 643 * 4);
  float* w_h1 = (float*)alloc((size_t)2048 * 256 * 4);
  float* w_h2 = (float*)alloc((size_t)2048 * 512 * 4);
  float* w_h3 = (float*)alloc((size_t)2048 * 1024 * 4);
  float* w_gm = (float*)alloc((size_t)BB * 1024 * 4);
  float* w_t1 = (float*)alloc((size_t)16 * 512 * 4);
  float* w_t2 = (float*)alloc((size_t)16 * 256 * 4);

  // ---- augmentation: chamfer angle search (WMMA) ----
  k_norms<<<(BB * NN + 255) / 256, 256, 0, stream>>>(pc, w_xx);
  hipMemsetAsync(w_cd, 0, (size_t)NANG * BB * 4, stream);
  k_chamfer<<<dim3(NN / 16, BB, NANG * 2), 32, 0, stream>>>(pc, w_xx, w_cd);
  k_select_theta<<<1, 32, 0, stream>>>(w_cd, w_th);
  k_minor_rot<<<(BB * NN + 255) / 256, 256, 0, stream>>>(pc, idxm, w_th, w_mrot, w_yy);

  // ---- Sinkhorn assignment + mixing ----
  hipMemsetAsync(w_f, 0, (size_t)BB * NN * 4, stream);
  hipMemsetAsync(w_g, 0, (size_t)BB * NN * 4, stream);
  for (int it = 0; it < SINKIT; ++it) {
    k_sink_f<<<dim3(NN / 256, BB), 256, 0, stream>>>(pc, w_xx, w_mrot, w_yy, w_g, w_f);
    k_sink_g<<<dim3(NN / 256, BB), 256, 0, stream>>>(pc, w_xx, w_mrot, w_yy, w_f, w_g);
  }
  k_assign<<<dim3(NN / 256, BB), 256, 0, stream>>>(pc, w_xx, w_mrot, w_yy, w_g, w_ass);
  k_mix<<<(BB * NN + 255) / 256, 256, 0, stream>>>(pc, idxm, mixr, w_ass, w_mix);

  // ---- SA1 (MSG) ----
  k_fps<<<BB, 256, 0, stream>>>(w_mix, NN, 512, w_fps1);
  k_gather<<<(BB * 512 + 255) / 256, 256, 0, stream>>>(w_mix, NN, w_fps1, 512, w_nx1);
  const float sa1_r[3] = {0.1f, 0.2f, 0.4f};
  const int sa1_ns[3] = {16, 32, 128};
  const int sa1_dims[3][4] = {{3, 32, 32, 64}, {3, 64, 64, 128}, {3, 64, 96, 128}};
  int coff = 0;
  for (int sI = 0; sI < 3; ++sI) {
    int base = sa1B + sI * 6;
    k_ball<<<(BB * 512 + 255) / 256, 256, 0, stream>>>(
        w_mix, NN, w_nx1, 512, sa1_r[sI] * sa1_r[sI], sa1_ns[sI], w_ball);
    k_group_mlp<<<BB * 512, GM_THREADS, 0, stream>>>(
        w_mix, (const float*)nullptr, 0, NN, w_nx1, 512, w_ball, sa1_ns[sI],
        F(base + 0), F(base + 1), sa1_dims[sI][1], F(base + 2), F(base + 3),
        sa1_dims[sI][2], F(base + 4), F(base + 5), sa1_dims[sI][3], w_ft1, 320,
        coff);
    coff += sa1_dims[sI][3];
  }

  // ---- SA2 (MSG) ----
  k_fps<<<BB, 256, 0, stream>>>(w_nx1, 512, 128, w_fps2);
  k_gather<<<(BB * 128 + 255) / 256, 256, 0, stream>>>(w_nx1, 512, w_fps2, 128, w_nx2);
  const float sa2_r[3] = {0.2f, 0.4f, 0.8f};
  const int sa2_ns[3] = {32, 64, 128};
  const int sa2_dims[3][4] = {
      {323, 64, 64, 128}, {323, 128, 128, 256}, {323, 128, 128, 256}};
  coff = 0;
  for (int sI = 0; sI < 3; ++sI) {
    int base = sa2B + sI * 6;
    k_ball<<<(BB * 128 + 255) / 256, 256, 0, stream>>>(
        w_nx1, 512, w_nx2, 128, sa2_r[sI] * sa2_r[sI], sa2_ns[sI], w_ball);
    k_group_mlp<<<BB * 128, GM_THREADS, 0, stream>>>(
        w_nx1, w_ft1, 320, 512, w_nx2, 128, w_ball, sa2_ns[sI], F(base + 0),
        F(base + 1), sa2_dims[sI][1], F(base + 2), F(base + 3), sa2_dims[sI][2],
        F(base + 4), F(base + 5), sa2_dims[sI][3], w_ft2, 640, coff);
    coff += sa2_dims[sI][3];
  }

  // ---- SA3 global MLP (WMMA GEMMs over 2048x643 rows) + max pool ----
  k_concat_sa3<<<(2048 * 643 + 255) / 256, 256, 0, stream>>>(w_nx2, w_ft2, w_s3i);
  k_gemm_wmma<<<dim3(256 / 16, 2048 / 16), 32, 0, stream>>>(
      w_s3i, F(sa3B + 0), F(sa3B + 1), w_h1, 2048, 643, 256, 1);
  k_gemm_wmma<<<dim3(512 / 16, 2048 / 16), 32, 0, stream>>>(
      w_h1, F(sa3B + 2), F(sa3B + 3), w_h2, 2048, 256, 512, 1);
  k_gemm_wmma<<<dim3(1024 / 16, 2048 / 16), 32, 0, stream>>>(
      w_h2, F(sa3B + 4), F(sa3B + 5), w_h3, 2048, 512, 1024, 1);
  k_maxpool_sa3<<<(BB * 1024 + 255) / 256, 256, 0, stream>>>(w_h3, w_gm);

  // ---- FC head (WMMA, M=16 = one tile) ----
  k_gemm_wmma<<<dim3(512 / 16, 1), 32, 0, stream>>>(w_gm, F(fcB + 0), F(fcB + 1),
                                                    w_t1, 16, 1024, 512, 1);
  k_gemm_wmma<<<dim3(256 / 16, 1), 32, 0, stream>>>(w_t1, F(fcB + 2), F(fcB + 3),
                                                    w_t2, 16, 512, 256, 1);
  k_gemm_wmma<<<dim3((40 + 15) / 16, 1), 32, 0, stream>>>(
      w_t2, F(fcB + 4), F(fcB + 5), (float*)d_out, 16, 256, 40, 0);
}